// Rotation90SymmetricPosEmbed_41120016892507
// MI455X (gfx1250) — compile-verified
//
#include <hip/hip_runtime.h>

// Rotation90SymmetricPosEmbed: out[b,t,:] = x[b,t,:] + pos_row(t)
// Pure HBM-streaming broadcast add. Roofline ~537 MB @ 23.3 TB/s => ~23 us.
// Strategy: closed-form (idx, rot) per grid cell (replicates the reference's
// sequential last-write-wins scatter), pos float4 held in registers, reused
// across an 8-batch strip with non-temporal 128-bit loads/stores so the
// 2.2 MB learned table stays resident in the 192 MB L2 across 64x reuse.

typedef float v4f __attribute__((ext_vector_type(4)));

#define CC      256          // quarter channel width
#define ROWF    1024         // 4*C floats per token row
#define TOKENS  1025         // 1 cls + 32*32 patches
#define BATCH   64
#define BPB     8            // batches per block (BATCH/BPB = gridDim.y)

// Enumeration index of upper-triangular position (i,j), i<=j, H=W=32:
// idx = sum_{r<i}(32-r) + (j-i) = 32*i - i*(i-1)/2 + (j-i)
__device__ __forceinline__ int tri_idx(int i, int j) {
    return i * 32 - ((i * (i - 1)) >> 1) + (j - i);
}

__global__ __launch_bounds__(256)
void pos_embed_add_kernel(const float* __restrict__ x,
                          const float* __restrict__ pos,   // [528, 1024]
                          const float* __restrict__ cls,   // [256]
                          float* __restrict__ out) {
    const int token = blockIdx.x;        // 0..1024
    const int bgrp  = blockIdx.y;        // 0..(BATCH/BPB - 1)
    const int c     = threadIdx.x << 2;  // channel 0..1023, float4 granularity

    // ---- compute this token's pos float4 (block-uniform control flow) ----
    v4f p;
    if (token == 0) {
        // cls row: quarter tiled 4x; c aligned to 4, 4 | 256 so the float4
        // never crosses a quarter boundary.
        p = *reinterpret_cast<const v4f*>(cls + (c & (CC - 1)));
    } else {
        const int cell = token - 1;
        const int a = cell >> 5;   // grid row
        const int b = cell & 31;   // grid col
        // Up to 4 writers of cell (a,b); last (sequential) write wins
        // == max over valid candidates of idx*4 + rot.
        int best = -1;
        if (a <= b)        { int k = (tri_idx(a,      b     ) << 2) | 0; best = best > k ? best : k; }
        if ((31 - b) <= a) { int k = (tri_idx(31 - b, a     ) << 2) | 1; best = best > k ? best : k; }
        if (b <= a)        { int k = (tri_idx(31 - a, 31 - b) << 2) | 2; best = best > k ? best : k; }
        if (b <= (31 - a)) { int k = (tri_idx(b,      31 - a) << 2) | 3; best = best > k ? best : k; }
        const int idx = best >> 2;
        const int rot = best & 3;
        // output quarter k reads learned quarter (rot+k)%4
        const int src_q = (rot + (c >> 8)) & 3;
        // regular-temporal load: keep the 2.2 MB table hot in L2 (64x reuse)
        p = *reinterpret_cast<const v4f*>(pos + (size_t)idx * ROWF
                                              + (src_q << 8) + (c & (CC - 1)));
    }

    // ---- stream BPB batches: NT b128 load -> add -> NT b128 store ----
    const size_t stride = (size_t)TOKENS * ROWF;          // one batch image
    size_t base = ((size_t)(bgrp * BPB) * TOKENS + (size_t)token) * ROWF + c;

    v4f v[BPB];
#pragma unroll
    for (int i = 0; i < BPB; ++i) {
        v[i] = __builtin_nontemporal_load(
                   reinterpret_cast<const v4f*>(x + base + (size_t)i * stride));
    }
#pragma unroll
    for (int i = 0; i < BPB; ++i) {
        v[i] += p;   // packed f32 adds
    }
#pragma unroll
    for (int i = 0; i < BPB; ++i) {
        __builtin_nontemporal_store(v[i],
                   reinterpret_cast<v4f*>(out + base + (size_t)i * stride));
    }
}

extern "C" void kernel_launch(void* const* d_in, const int* in_sizes, int n_in,
                              void* d_out, int out_size, void* d_ws, size_t ws_size,
                              hipStream_t stream) {
    const float* x   = (const float*)d_in[0];  // [64, 1025, 1024]
    const float* pos = (const float*)d_in[1];  // [1, 528, 1024]
    const float* cls = (const float*)d_in[2];  // [1, 1, 256]
    float* out = (float*)d_out;                // [64, 1025, 1024]

    dim3 grid(TOKENS, BATCH / BPB);
    dim3 block(256);
    pos_embed_add_kernel<<<grid, block, 0, stream>>>(x, pos, cls, out);
}